// NeuralG_8564164788386
// MI455X (gfx1250) — compile-verified
//
#include <hip/hip_runtime.h>
#include <cstddef>

// Problem constants (match reference)
#define HD   16          // hidden dim
#define NP   304         // param count = 16*16 + 3*16
#define TT   256         // timesteps
#define HH0  16          // w_hh params start
#define HH1  272         // w_hh params end
#define BH0  272         // b_h start
#define OUT0 288         // w_out start
#define THRESH 1e-4f

typedef float v2f __attribute__((ext_vector_type(2)));
typedef float v8f __attribute__((ext_vector_type(8)));

__device__ __forceinline__ int   f2i(float x) { return __builtin_bit_cast(int, x); }
__device__ __forceinline__ float i2f(int x)   { return __builtin_bit_cast(float, x); }

#if __has_builtin(__builtin_amdgcn_permlane16_swap)
#define HAVE_PLSWAP 1
typedef int v2i __attribute__((ext_vector_type(2)));
#endif

// ---------------------------------------------------------------------------
// A-operand for V_WMMA_F32_16X16X4_F32: A = w_hh[:, 4s:4s+4] (16x4 f32).
// Layout (ISA 7.12.2): lanes hold M = lane&15; VGPR0 holds K=4s (lo half) /
// K=4s+2 (hi half); VGPR1 holds K=4s+1 / K=4s+3.
// ---------------------------------------------------------------------------
__device__ __forceinline__ void load_A_whh(const float* __restrict__ fp,
                                           int lane, v2f A[4]) {
  const int m = lane & 15;
  const int half = lane >> 4;
#pragma unroll
  for (int s = 0; s < 4; ++s) {
    A[s].x = fp[HD + m * HD + 4 * s + 0 + 2 * half];
    A[s].y = fp[HD + m * HD + 4 * s + 1 + 2 * half];
  }
}

// ---------------------------------------------------------------------------
// Repack a 16x16 f32 tile held in C/D layout (c[r]: row r lo-half, row r+8
// hi-half; col = lane&15) into four 4x16 B operands (rows 4s..4s+3).
// v_permlane16_swap_b32: (a,b) -> ({a.lo,b.lo}, {a.hi,b.hi}) — each swap
// yields TWO B regs, pure VALU (no ds_bpermute / DScnt waits).
// ---------------------------------------------------------------------------
#ifdef HAVE_PLSWAP
__device__ __forceinline__ void make_B(const float c[8], bool /*lt*/, v2f B[4]) {
  v2i r02 = __builtin_amdgcn_permlane16_swap(f2i(c[0]), f2i(c[2]), false, false);
  v2i r13 = __builtin_amdgcn_permlane16_swap(f2i(c[1]), f2i(c[3]), false, false);
  v2i r46 = __builtin_amdgcn_permlane16_swap(f2i(c[4]), f2i(c[6]), false, false);
  v2i r57 = __builtin_amdgcn_permlane16_swap(f2i(c[5]), f2i(c[7]), false, false);
  B[0].x = i2f(r02.x); B[0].y = i2f(r13.x);   // rows 0,1 / 2,3
  B[1].x = i2f(r46.x); B[1].y = i2f(r57.x);   // rows 4,5 / 6,7
  B[2].x = i2f(r02.y); B[2].y = i2f(r13.y);   // rows 8,9 / 10,11
  B[3].x = i2f(r46.y); B[3].y = i2f(r57.y);   // rows 12,13 / 14,15
}
#else
__device__ __forceinline__ void make_B(const float c[8], bool lt, v2f B[4]) {
  float sh[8];
#pragma unroll
  for (int r = 0; r < 8; ++r) sh[r] = __shfl_xor(c[r], 16, 32);
  B[0].x = lt ? c[0] : sh[2];  B[0].y = lt ? c[1] : sh[3];
  B[1].x = lt ? c[4] : sh[6];  B[1].y = lt ? c[5] : sh[7];
  B[2].x = lt ? sh[0] : c[2];  B[2].y = lt ? sh[1] : c[3];
  B[3].x = lt ? sh[4] : c[6];  B[3].y = lt ? sh[5] : c[7];
}
#endif

// Half-wave sum for the w_out reductions: result valid on lanes 0..15
// (the only lanes that store). permlane16_swap(p,p).y holds, on lanes 0-15,
// the value of p from lane+16 — one VALU op instead of ds_bpermute.
__device__ __forceinline__ float half_sum16(float p) {
#ifdef HAVE_PLSWAP
  v2i r = __builtin_amdgcn_permlane16_swap(f2i(p), f2i(p), false, false);
  return p + i2f(r.y);
#else
  return p + __shfl_xor(p, 16, 32);
#endif
}

// out = w_hh(16x16) @ tile(16x16) via 4 chained f32 WMMAs (K-slices of 4).
__device__ __forceinline__ void wmma_W_x_tile(const v2f A[4], const float c[8],
                                              bool lt, float out[8]) {
  v2f B[4];
  make_B(c, lt, B);
  v8f acc = {0.f, 0.f, 0.f, 0.f, 0.f, 0.f, 0.f, 0.f};
#pragma unroll
  for (int s = 0; s < 4; ++s)
    acc = __builtin_amdgcn_wmma_f32_16x16x4_f32(
        false, A[s], false, B[s], (short)0, acc, false, false);
#pragma unroll
  for (int r = 0; r < 8; ++r) out[r] = acc[r];
}

// Add v into row `ti` (per-lane column) of a C-layout tile; ti == -1 -> nop.
__device__ __forceinline__ void row_add(float c[8], int half, int ti, float v) {
#pragma unroll
  for (int r = 0; r < 8; ++r) c[r] += (ti == r + 8 * half) ? v : 0.0f;
}

// ===========================================================================
// Kernel 1: serial h / decay / reset recurrence (1 wave, lanes 0..15 = units)
// Cold path (~256 tiny iterations) — shuffles here are irrelevant to runtime.
// ===========================================================================
__global__ void k1_state(const float* __restrict__ X,
                         const float* __restrict__ fp,
                         float* __restrict__ h_all, float* __restrict__ sp_all,
                         float* __restrict__ sd_all, float* __restrict__ reset_all,
                         float* __restrict__ Zout) {
  const int lane = threadIdx.x;
  const bool act = lane < HD;
  const float w_ih = act ? fp[lane] : 0.f;
  const float b_h  = act ? fp[BH0 + lane] : 0.f;
  const float w_o  = act ? fp[OUT0 + lane] : 0.f;
  float Wr[HD];
#pragma unroll
  for (int j = 0; j < HD; ++j) Wr[j] = act ? fp[HD + lane * HD + j] : 0.f;

  // ||w_hh||_F  (xor offsets 1..8 stay within each 16-lane half)
  float nrm = 0.f;
#pragma unroll
  for (int j = 0; j < HD; ++j) nrm += Wr[j] * Wr[j];
#pragma unroll
  for (int off = 1; off < 16; off <<= 1) nrm += __shfl_xor(nrm, off, 32);
  const float wnorm = sqrtf(nrm);

  float h = 0.f, decay = 1.f;
  for (int t = 0; t < TT; ++t) {
    const float x = X[t];
    const bool reset = decay < THRESH;
    if (reset) decay = 1.f;
    if (lane == 0) reset_all[t] = reset ? 1.f : 0.f;

    // z = w_out . h   (uses carry h)
    float p = w_o * h;
#pragma unroll
    for (int off = 1; off < 16; off <<= 1) p += __shfl_xor(p, off, 32);
    if (lane == 0) Zout[t] = p;
    if (act) h_all[t * HD + lane] = h;

    // a = x*w_ih + w_hh@h + b_h ; h_new = tanh(a)
    float a = x * w_ih + b_h;
#pragma unroll
    for (int j = 0; j < HD; ++j) a += Wr[j] * __shfl(h, j, 32);
    const float hn = tanhf(a);
    const float sp = 1.f - hn * hn;
    const float sd = -2.f * hn * sp;
    if (act) { sp_all[t * HD + lane] = sp; sd_all[t * HD + lane] = sd; }

    float m = act ? fabsf(sp) : 0.f;
#pragma unroll
    for (int off = 1; off < 16; off <<= 1) m = fmaxf(m, __shfl_xor(m, off, 32));
    decay *= m * wnorm;
    h = hn;
  }
}

// ===========================================================================
// Kernel 2: Jh recurrence. 19 independent waves; wave lg owns columns
// [16*lg, 16*lg+16) of Jh (16 x 304) for all t. Column-local recurrence:
//   Ja = w_hh@Jh + structured adds ; Jh' = sp (.) Ja ; Jz = w_out@Jh (+h).
// Jh/Ja are persisted TRANSPOSED as [t][col][i]: each lane's 8 row-values
// (i = 8*half..8*half+7, matching the C-layout registers) are one contiguous
// 32B run -> 2x b128 per array instead of 8x b32 scatter.
// ===========================================================================
__global__ void k2_jac(const float* __restrict__ X,
                       const float* __restrict__ fp,
                       const float* __restrict__ h_all,
                       const float* __restrict__ sp_all,
                       const float* __restrict__ reset_all,
                       float* __restrict__ JhT, float* __restrict__ JaT,
                       float* __restrict__ Jout) {
  const int lane = threadIdx.x;
  const int half = lane >> 4;
  const int cix  = lane & 15;
  const bool lt  = (half == 0);
  const int l0   = blockIdx.x * 16;
  const int col  = l0 + cix;

  v2f A[4];
  load_A_whh(fp, lane, A);
  float wsel[8];
#pragma unroll
  for (int r = 0; r < 8; ++r) wsel[r] = fp[OUT0 + r + 8 * half];

  float jh[8];
#pragma unroll
  for (int r = 0; r < 8; ++r) jh[r] = 0.f;

  for (int t = 0; t < TT; ++t) {
    const float x = X[t];
    if (reset_all[t] != 0.f) {
#pragma unroll
      for (int r = 0; r < 8; ++r) jh[r] = 0.f;
    }
    // persist carry-in (post-reset) Jh_t, transposed [t][col][i]
    {
      float4* p = (float4*)&JhT[((size_t)(t * NP + col)) * HD + 8 * half];
      p[0] = make_float4(jh[0], jh[1], jh[2], jh[3]);
      p[1] = make_float4(jh[4], jh[5], jh[6], jh[7]);
    }

    // Jz[t] = w_out @ Jh_t  (+ h on the w_out columns)
    float p = 0.f;
#pragma unroll
    for (int r = 0; r < 8; ++r) p += wsel[r] * jh[r];
    p = half_sum16(p);
    if (lane < 16) {
      float jz = p;
      if (col >= OUT0) jz += h_all[t * HD + (col - OUT0)];
      Jout[t * NP + col] = jz;
    }

    // Ja = w_hh @ Jh_t  (4x v_wmma_f32_16x16x4_f32)
    float ja[8];
    wmma_W_x_tile(A, jh, lt, ja);

    // structured adds (each 16-col strip lies in exactly one range)
    int ti = -1; float v = 0.f;
    if (col < HH0)        { ti = col;              v = x; }                 // x*I
    else if (col < HH1)   { ti = (col - HH0) >> 4; v = h_all[t * HD + ((col - HH0) & 15)]; } // +h[j]
    else if (col < OUT0)  { ti = col - BH0;        v = 1.f; }               // +I
    row_add(ja, half, ti, v);

    {
      float4* q = (float4*)&JaT[((size_t)(t * NP + col)) * HD + 8 * half];
      q[0] = make_float4(ja[0], ja[1], ja[2], ja[3]);
      q[1] = make_float4(ja[4], ja[5], ja[6], ja[7]);
    }

    // Jh' = sp (.) Ja
    {
      const float4* sp4 = (const float4*)&sp_all[t * HD + 8 * half];
      const float4 s0 = sp4[0], s1 = sp4[1];
      jh[0] = s0.x * ja[0]; jh[1] = s0.y * ja[1];
      jh[2] = s0.z * ja[2]; jh[3] = s0.w * ja[3];
      jh[4] = s1.x * ja[4]; jh[5] = s1.y * ja[5];
      jh[6] = s1.z * ja[6]; jh[7] = s1.w * ja[7];
    }
  }
}

// ===========================================================================
// Kernel 3: Hh recurrence + Hm output. 5776 waves; wave w owns the 16-entry
// column group (k = w/19, l in [16*(w%19), +16)) of Hh (16 x 304 x 304),
// kept resident in one 8-VGPR C-layout tile across all 256 steps.
// ===========================================================================
__global__ void k3_hess(const float* __restrict__ fp,
                        const float* __restrict__ sp_all,
                        const float* __restrict__ sd_all,
                        const float* __restrict__ reset_all,
                        const float* __restrict__ JhT,
                        const float* __restrict__ JaT,
                        float* __restrict__ Hm) {
  const int wid = blockIdx.x * (blockDim.x >> 5) + (threadIdx.x >> 5);
  if (wid >= NP * 19) return;  // exact fit; wave-uniform
  const int k  = wid / 19;
  const int l0 = (wid % 19) * 16;
  const int lane = threadIdx.x & 31;
  const int half = lane >> 4;
  const int cix  = lane & 15;
  const bool lt  = (half == 0);
  const int l    = l0 + cix;

  v2f A[4];
  load_A_whh(fp, lane, A);
  float wsel[8];
#pragma unroll
  for (int r = 0; r < 8; ++r) wsel[r] = fp[OUT0 + r + 8 * half];

  // D-term row metadata (uniform per wave): k in w_hh block?
  const bool kInHH  = (k >= HH0) && (k < HH1);
  const int  kRow   = kInHH ? ((k - HH0) >> 4) : -1;   // i*
  const int  kSub   = kInHH ? ((k - HH0) & 15) : 0;    // j*
  // D^T-term column metadata (per lane): l in w_hh block?
  const int  lRow   = (l >= HH0 && l < HH1) ? ((l - HH0) >> 4) : -1;
  const int  lSub   = (l >= HH0 && l < HH1) ? ((l - HH0) & 15) : 0;

  float hh[8];
#pragma unroll
  for (int r = 0; r < 8; ++r) hh[r] = 0.f;

  for (int t = 0; t < TT; ++t) {
    const size_t tb = (size_t)t * NP;  // column base into JhT / JaT
    if (reset_all[t] != 0.f) {
#pragma unroll
      for (int r = 0; r < 8; ++r) hh[r] = 0.f;
    }

    // ---- Hm[t][k][l0..l0+15] = w_out @ Hh_t + Dz + Dz^T ----
    float p = 0.f;
#pragma unroll
    for (int r = 0; r < 8; ++r) p += wsel[r] * hh[r];
    p = half_sum16(p);
    if (lane < 16) {
      float hm = p;
      if (k >= OUT0) hm += JhT[(tb + l) * HD + (k - OUT0)];
      if (l >= OUT0) hm += JhT[(tb + k) * HD + (l - OUT0)];
      Hm[(size_t)t * NP * NP + (size_t)k * NP + l] = hm;
    }

    // ---- Ha = w_hh @ Hh_t  (4x v_wmma_f32_16x16x4_f32) ----
    float ha[8];
    wmma_W_x_tile(A, hh, lt, ha);

    // D:  Ha[i*, l] += Jh_t[j*, l]           (row uniform, value per-lane)
    if (kInHH) row_add(ha, half, kRow, JhT[(tb + l) * HD + kSub]);
    // D^T: Ha[i_c, l] += Jh_t[j_c, k]        (row and value per-lane; the
    // gather stays inside one 64B line in the [t][col][i] layout)
    {
      const float v = (lRow >= 0) ? JhT[(tb + k) * HD + lSub] : 0.f;
      row_add(ha, half, lRow, v);
    }

    // ---- Hh' = sp (.) Ha + sd (.) Ja(:,k) Ja(:,l) ----
    {
      const float4* sp4 = (const float4*)&sp_all[t * HD + 8 * half];
      const float4* sd4 = (const float4*)&sd_all[t * HD + 8 * half];
      const float4* jk4 = (const float4*)&JaT[(tb + k) * HD + 8 * half];
      const float4* jl4 = (const float4*)&JaT[(tb + l) * HD + 8 * half];
      const float4 s0 = sp4[0], s1 = sp4[1];
      const float4 d0 = sd4[0], d1 = sd4[1];
      const float4 k0 = jk4[0], k1 = jk4[1];
      const float4 l0v = jl4[0], l1v = jl4[1];
      hh[0] = s0.x * ha[0] + d0.x * k0.x * l0v.x;
      hh[1] = s0.y * ha[1] + d0.y * k0.y * l0v.y;
      hh[2] = s0.z * ha[2] + d0.z * k0.z * l0v.z;
      hh[3] = s0.w * ha[3] + d0.w * k0.w * l0v.w;
      hh[4] = s1.x * ha[4] + d1.x * k1.x * l1v.x;
      hh[5] = s1.y * ha[5] + d1.y * k1.y * l1v.y;
      hh[6] = s1.z * ha[6] + d1.z * k1.z * l1v.z;
      hh[7] = s1.w * ha[7] + d1.w * k1.w * l1v.w;
    }

    if (t + 1 < TT) {  // warm L2/L0 for next step's shared columns
      __builtin_prefetch(&JaT[((size_t)(t + 1) * NP + l) * HD], 0, 0);
      __builtin_prefetch(&JhT[((size_t)(t + 1) * NP + k) * HD], 0, 0);
    }
  }
}

// ===========================================================================
extern "C" void kernel_launch(void* const* d_in, const int* in_sizes, int n_in,
                              void* d_out, int out_size, void* d_ws, size_t ws_size,
                              hipStream_t stream) {
  const float* X  = (const float*)d_in[0];   // (256,)
  const float* fp = (const float*)d_in[1];   // (304,)
  float* out = (float*)d_out;
  float* Z   = out;                          // 256
  float* J   = out + TT;                     // 256*304
  float* Hm  = out + TT + (size_t)TT * NP;   // 256*304*304

  float* ws        = (float*)d_ws;           // ~10 MB used
  float* h_all     = ws;                               // 256*16
  float* sp_all    = h_all    + (size_t)TT * HD;       // 256*16 (16B aligned)
  float* sd_all    = sp_all   + (size_t)TT * HD;       // 256*16
  float* reset_all = sd_all   + (size_t)TT * HD;       // 256
  float* JhT       = reset_all + TT;                   // 256*304*16, [t][col][i]
  float* JaT       = JhT      + (size_t)TT * NP * HD;  // 256*304*16, [t][col][i]

  k1_state<<<1, 32, 0, stream>>>(X, fp, h_all, sp_all, sd_all, reset_all, Z);
  k2_jac<<<19, 32, 0, stream>>>(X, fp, h_all, sp_all, reset_all, JhT, JaT, J);
  k3_hess<<<(NP * 19) / 8, 256, 0, stream>>>(fp, sp_all, sd_all, reset_all,
                                             JhT, JaT, Hm);
}